// GatedGCNLayer_43164421325632
// MI455X (gfx1250) — compile-verified
//
#include <hip/hip_runtime.h>
#include <hip/hip_bf16.h>

typedef __attribute__((ext_vector_type(2))) float v2f;
typedef __attribute__((ext_vector_type(8))) float v8f;

#define D        128
#define NNODE    256
#define ROWS     1024        // B*N
#define EDGECNT  262144.0f   // B*N*N
#define NODECNT  1024.0f
#define BN_EPS   1e-5f

// workspace float offsets
#define WS_AH     0
#define WS_BH     131072
#define WS_UH     262144
#define WS_VH     393216
#define WS_AGG    524288
#define WS_HACT   655360
#define WS_ESUM   786432
#define WS_ESQ    786560
#define WS_NSUM   786688
#define WS_NSQ    786816
#define WS_ESCALE 786944
#define WS_ESHIFT 787072
#define WS_NSCALE 787200
#define WS_NSHIFT 787328

// LDS layout for edge kernel (floats)
#define LDS_WC    0                       // 128*132
#define LDS_E     (128*132)               // 8 waves * 2 bufs * 16*132
#define LDS_SUM   (LDS_E + 8*2*16*132)
#define LDS_SQ    (LDS_SUM + 128)
#define LDS_AGG   (LDS_SQ + 128)
#define LDS_TOTAL (LDS_AGG + 128)

// async copy of 16B global -> LDS, tracked by ASYNCcnt
__device__ __forceinline__ void async_b128(uint32_t lds_bytes, const float* g) {
    asm volatile("global_load_async_to_lds_b128 %0, %1, off"
                 :: "v"(lds_bytes), "v"(g) : "memory");
}

__global__ void init_stats_kernel(float* acc) {
    acc[threadIdx.x] = 0.0f;   // 512 floats: eSum,eSq,nSum,nSq
}

// Ah/Bh/Uh/Vh = h @ W^T via v_wmma_f32_16x16x4_f32. grid=(64,2,4), block=128 (4 waves)
__global__ void node_gemm_kernel(const float* __restrict__ h,
                                 const float* __restrict__ wA, const float* __restrict__ wB,
                                 const float* __restrict__ wU, const float* __restrict__ wV,
                                 float* __restrict__ ws) {
    const int tid  = threadIdx.x;
    const int wave = tid >> 5, lane = tid & 31;
    const int n16  = lane & 15, hi = lane >> 4;
    const int m0   = blockIdx.x * 16;
    const int n0   = (blockIdx.y * 4 + wave) * 16;
    const int z    = blockIdx.z;
    const float* W = (z == 0) ? wA : (z == 1) ? wB : (z == 2) ? wU : wV;
    float* O       = ws + (size_t)z * 131072;

    v8f c = {0.f, 0.f, 0.f, 0.f, 0.f, 0.f, 0.f, 0.f};
    const float* arow = h + (size_t)(m0 + n16) * D;
    const float* brow = W + (size_t)(n0 + n16) * D;
#pragma unroll
    for (int kk = 0; kk < 32; ++kk) {
        const int col = kk * 4 + hi * 2;
        v2f a;  a.x  = arow[col]; a.y  = arow[col + 1];
        v2f bb; bb.x = brow[col]; bb.y = brow[col + 1];
        c = __builtin_amdgcn_wmma_f32_16x16x4_f32(false, a, false, bb, (short)0, c, false, false);
    }
#pragma unroll
    for (int rr = 0; rr < 8; ++rr) {
        const int m = rr + 8 * hi;
        O[(size_t)(m0 + m) * D + n0 + n16] = c[rr];
    }
}

// One workgroup per (b,i): G = e[b,i,:,:] @ wC^T + Ah_i + Bh_j ; write gate_input,
// accumulate edge-BN partial sums, and agg[b,i,:] = sum_j sigmoid(G)*Vh[b,j,:].
// grid=1024, block=256 (8 waves), dynamic LDS = LDS_TOTAL floats.
// Staging via global_load_async_to_lds_b128, double-buffered per wave.
__global__ void edge_kernel(const float* __restrict__ e,
                            const float* __restrict__ Ah, const float* __restrict__ Bh,
                            const float* __restrict__ Vh, const float* __restrict__ wC,
                            float* __restrict__ gate_out, float* __restrict__ agg,
                            float* __restrict__ eSum, float* __restrict__ eSq) {
    extern __shared__ float smem[];
    float* sWC  = smem + LDS_WC;
    float* sSum = smem + LDS_SUM;
    float* sSq  = smem + LDS_SQ;
    float* sAgg = smem + LDS_AGG;

    const int r    = blockIdx.x;       // b*256 + i
    const int b    = r >> 8;
    const int tid  = threadIdx.x;
    const int wave = tid >> 5, lane = tid & 31;
    const int n16  = lane & 15, hi = lane >> 4;
    const size_t erow = (size_t)r * NNODE;

    // 1) issue async staging of wC (16 instr/wave): rows padded to 132 floats
    {
        const uint32_t swc_base = (uint32_t)(uintptr_t)sWC;
        const int base = tid * 64;
#pragma unroll
        for (int q = 0; q < 16; ++q) {
            const int idx = base + q * 4;
            const int row = idx >> 7, col = idx & 127;
            async_b128(swc_base + (uint32_t)(row * 132 + col) * 4u, wC + idx);
        }
    }
    // 2) issue async staging of this wave's two 16-row e slices (16 instr/wave each)
#pragma unroll
    for (int buf = 0; buf < 2; ++buf) {
        const int j0 = (wave * 2 + buf) * 16;
        const uint32_t se_base =
            (uint32_t)(uintptr_t)(smem + LDS_E + (wave * 2 + buf) * (16 * 132));
        const int base = lane * 64;
#pragma unroll
        for (int q = 0; q < 16; ++q) {
            const int idx = base + q * 4;
            const int row = idx >> 7, col = idx & 127;
            async_b128(se_base + (uint32_t)(row * 132 + col) * 4u,
                       e + (erow + j0 + row) * D + col);
        }
    }
    if (tid < 128) { sSum[tid] = 0.f; sSq[tid] = 0.f; sAgg[tid] = 0.f; }

    // wC done when own ASYNCcnt <= 32 (async loads complete in order); then all-wave barrier
    asm volatile("s_wait_asynccnt 0x20" ::: "memory");
    __syncthreads();

#pragma unroll
    for (int buf = 0; buf < 2; ++buf) {
        if (buf == 0) asm volatile("s_wait_asynccnt 0x10" ::: "memory");
        else          asm volatile("s_wait_asynccnt 0x0"  ::: "memory");

        const int j0 = (wave * 2 + buf) * 16;
        const float* sEwB = smem + LDS_E + (wave * 2 + buf) * (16 * 132);

        // 8 independent accumulators: one per 16-wide k-tile
        v8f c[8];
#pragma unroll
        for (int kt = 0; kt < 8; ++kt)
#pragma unroll
            for (int rr = 0; rr < 8; ++rr) c[kt][rr] = 0.f;

        const float* arow = sEwB + n16 * 132;
#pragma unroll 4
        for (int kk = 0; kk < 32; ++kk) {
            const int col = kk * 4 + hi * 2;
            v2f a; a.x = arow[col]; a.y = arow[col + 1];
#pragma unroll
            for (int kt = 0; kt < 8; ++kt) {
                const float* pb = sWC + (kt * 16 + n16) * 132 + col;
                v2f bb; bb.x = pb[0]; bb.y = pb[1];
                c[kt] = __builtin_amdgcn_wmma_f32_16x16x4_f32(
                    false, a, false, bb, (short)0, c[kt], false, false);
            }
        }

        // epilogue: add Ah_i + Bh_j, emit gate_input, BN partials, agg partials
#pragma unroll
        for (int kt = 0; kt < 8; ++kt) {
            const int k0 = kt * 16;
            const float ah = Ah[(size_t)r * D + k0 + n16];
            float sacc = 0.f, qacc = 0.f, aacc = 0.f;
#pragma unroll
            for (int rr = 0; rr < 8; ++rr) {
                const int m = rr + 8 * hi;
                const int j = j0 + m;
                const float g = c[kt][rr] + ah + Bh[((size_t)b * NNODE + j) * D + k0 + n16];
                gate_out[(erow + j) * D + k0 + n16] = g;
                sacc += g;
                qacc += g * g;
                const float sig = 1.f / (1.f + __expf(-g));
                aacc += sig * Vh[((size_t)b * NNODE + j) * D + k0 + n16];
            }
            atomicAdd(&sSum[k0 + n16], sacc);
            atomicAdd(&sSq[k0 + n16], qacc);
            atomicAdd(&sAgg[k0 + n16], aacc);
        }
    }
    __syncthreads();
    if (tid < 128) {
        atomicAdd(&eSum[tid], sSum[tid]);
        atomicAdd(&eSq[tid], sSq[tid]);
        agg[(size_t)r * D + tid] = sAgg[tid];
    }
}

// h_act = relu(Uh + agg), node BN partial sums. grid=128, block=128
__global__ void node_act_kernel(const float* __restrict__ Uh, const float* __restrict__ agg,
                                float* __restrict__ hact, float* nSum, float* nSq) {
    const int d = threadIdx.x;
    const int row0 = blockIdx.x * 8;
    float s = 0.f, q = 0.f;
#pragma unroll
    for (int k = 0; k < 8; ++k) {
        const size_t idx = (size_t)(row0 + k) * D + d;
        float v = Uh[idx] + agg[idx];
        v = fmaxf(v, 0.f);
        hact[idx] = v;
        s += v; q += v * v;
    }
    atomicAdd(&nSum[d], s);
    atomicAdd(&nSq[d], q);
}

// fold sums into per-channel scale/shift. grid=1, block=128
__global__ void stats_kernel(const float* eSum, const float* eSq,
                             const float* nSum, const float* nSq,
                             const float* g_node, const float* b_node,
                             const float* g_edge, const float* b_edge,
                             float* eScale, float* eShift, float* nScale, float* nShift) {
    const int d = threadIdx.x;
    {
        const float m = eSum[d] * (1.f / EDGECNT);
        const float var = eSq[d] * (1.f / EDGECNT) - m * m;
        const float s = g_edge[d] * rsqrtf(var + BN_EPS);
        eScale[d] = s; eShift[d] = b_edge[d] - m * s;
    }
    {
        const float m = nSum[d] * (1.f / NODECNT);
        const float var = nSq[d] * (1.f / NODECNT) - m * m;
        const float s = g_node[d] * rsqrtf(var + BN_EPS);
        nScale[d] = s; nShift[d] = b_node[d] - m * s;
    }
}

// h_new = BN(h_act) + h. grid=512, block=256
__global__ void node_out_kernel(const float* __restrict__ hact, const float* __restrict__ h,
                                const float* __restrict__ nScale, const float* __restrict__ nShift,
                                float* __restrict__ out) {
    const int idx = blockIdx.x * 256 + threadIdx.x;
    const int d = idx & 127;
    out[idx] = hact[idx] * nScale[d] + nShift[d] + h[idx];
}

// e_new = relu(BN(gate_input)) + e, in place on d_out e-region. grid=32768, block=256
__global__ void edge_out_kernel(float* __restrict__ eout, const float* __restrict__ e,
                                const float* __restrict__ eScale, const float* __restrict__ eShift) {
    const size_t vi = (size_t)blockIdx.x * 256 + threadIdx.x;
    const size_t base = vi * 4;
    const int d = (int)(base & 127);
    const float4 g  = *(const float4*)(eout + base);
    const float4 ev = *(const float4*)(e + base);
    const float4 sc = *(const float4*)(eScale + d);
    const float4 sh = *(const float4*)(eShift + d);
    float4 o;
    o.x = fmaxf(g.x * sc.x + sh.x, 0.f) + ev.x;
    o.y = fmaxf(g.y * sc.y + sh.y, 0.f) + ev.y;
    o.z = fmaxf(g.z * sc.z + sh.z, 0.f) + ev.z;
    o.w = fmaxf(g.w * sc.w + sh.w, 0.f) + ev.w;
    *(float4*)(eout + base) = o;
}

extern "C" void kernel_launch(void* const* d_in, const int* in_sizes, int n_in,
                              void* d_out, int out_size, void* d_ws, size_t ws_size,
                              hipStream_t stream) {
    const float* h      = (const float*)d_in[0];
    const float* e      = (const float*)d_in[1];
    const float* wA     = (const float*)d_in[2];
    const float* wB     = (const float*)d_in[3];
    const float* wU     = (const float*)d_in[4];
    const float* wV     = (const float*)d_in[5];
    const float* wC     = (const float*)d_in[6];
    const float* g_node = (const float*)d_in[7];
    const float* b_node = (const float*)d_in[8];
    const float* g_edge = (const float*)d_in[9];
    const float* b_edge = (const float*)d_in[10];

    float* out  = (float*)d_out;
    float* hout = out;                 // 131072 floats
    float* eout = out + 131072;        // 33554432 floats (staged gate_input, then e_new)
    float* ws   = (float*)d_ws;

    init_stats_kernel<<<1, 512, 0, stream>>>(ws + WS_ESUM);

    node_gemm_kernel<<<dim3(64, 2, 4), 128, 0, stream>>>(h, wA, wB, wU, wV, ws);

    const size_t ldsBytes = (size_t)LDS_TOTAL * sizeof(float);
    edge_kernel<<<1024, 256, ldsBytes, stream>>>(e, ws + WS_AH, ws + WS_BH, ws + WS_VH,
                                                 wC, eout, ws + WS_AGG,
                                                 ws + WS_ESUM, ws + WS_ESQ);

    node_act_kernel<<<128, 128, 0, stream>>>(ws + WS_UH, ws + WS_AGG, ws + WS_HACT,
                                             ws + WS_NSUM, ws + WS_NSQ);

    stats_kernel<<<1, 128, 0, stream>>>(ws + WS_ESUM, ws + WS_ESQ, ws + WS_NSUM, ws + WS_NSQ,
                                        g_node, b_node, g_edge, b_edge,
                                        ws + WS_ESCALE, ws + WS_ESHIFT,
                                        ws + WS_NSCALE, ws + WS_NSHIFT);

    node_out_kernel<<<512, 256, 0, stream>>>(ws + WS_HACT, h, ws + WS_NSCALE, ws + WS_NSHIFT, hout);

    edge_out_kernel<<<32768, 256, 0, stream>>>(eout, e, ws + WS_ESCALE, ws + WS_ESHIFT);
}